// TaskAlignedAssigner_20418274525580
// MI455X (gfx1250) — compile-verified
//
#include <hip/hip_runtime.h>

// Problem constants (match reference)
#define PP 65536
#define CC 80
#define NN 256
#define TOPK 13
#define EPSV 1e-9f
#define TILE 256                 // priors per stage-1 block
#define NTILES (PP / TILE)       // 256
#define SROW (TILE + 1)          // padded LDS row stride (bank-conflict free column reads)
#define CAND (NTILES * TOPK)     // 3328 candidates per gt in stage 2

typedef __attribute__((ext_vector_type(2))) float v2f;
typedef __attribute__((ext_vector_type(8))) float v8f;

// ---------------------------------------------------------------------------
// CDNA5 async global->LDS copies (ASYNCcnt-tracked, no VGPR round trip).
// GV mode: vdst = LDS byte offset (low 32 bits of flat LDS address),
//          vaddr = 64-bit global address.
// ---------------------------------------------------------------------------
__device__ __forceinline__ void async_g2lds_b128(void* lds_dst, const void* gsrc) {
    unsigned int l = (unsigned int)(unsigned long long)lds_dst;
    unsigned long long g = (unsigned long long)gsrc;
    asm volatile("global_load_async_to_lds_b128 %0, %1, off"
                 :: "v"(l), "v"(g) : "memory");
}
__device__ __forceinline__ void async_g2lds_b32(void* lds_dst, const void* gsrc) {
    unsigned int l = (unsigned int)(unsigned long long)lds_dst;
    unsigned long long g = (unsigned long long)gsrc;
    asm volatile("global_load_async_to_lds_b32 %0, %1, off"
                 :: "v"(l), "v"(g) : "memory");
}
__device__ __forceinline__ void async_wait0() {
    asm volatile("s_wait_asynccnt 0x0" ::: "memory");
}

// ---------------------------------------------------------------------------
// stage 0: init per-prior scatter state (ws is not re-poisoned between runs)
// ---------------------------------------------------------------------------
__global__ __launch_bounds__(256) void stage0_init(int* __restrict__ counts,
                                                   int* __restrict__ firstg) {
    int i = blockIdx.x * 256 + threadIdx.x;
    counts[i] = 0;
    firstg[i] = 0x7fffffff;
}

// ---------------------------------------------------------------------------
// stage 1: per prior-tile.
//   scores tile (256 gts x 256 priors) = onehot(labels)[256x80] x cls_pred^T
//   via V_WMMA_F32_16X16X4_F32, then fused IoU + align + per-gt top-13.
// ---------------------------------------------------------------------------
__global__ __launch_bounds__(256) void stage1_topk(
        const float* __restrict__ cls_pred,   // [P][80]
        const float* __restrict__ bbox_pred,  // [P][4]
        const int*   __restrict__ label_gt,   // [N]
        const float* __restrict__ bbox_gt,    // [N][4]
        float* __restrict__ candV,            // [N][NTILES][TOPK]
        int*   __restrict__ candI) {
    extern __shared__ char smem_raw[];
    float4* sGt  = (float4*)smem_raw;            // 256 gt boxes
    float4* sPb  = sGt + NN;                     // 256 pred boxes (this tile)
    float*  sS   = (float*)(sPb + TILE);         // score tile [256][SROW]
    int*    sLbl = (int*)(sS + NN * SROW);       // 256 labels

    const int tid = threadIdx.x;
    const int p0  = blockIdx.x * TILE;

    // async staging: boxes + labels straight into LDS
    async_g2lds_b128(&sGt[tid], bbox_gt + (size_t)tid * 4);
    async_g2lds_b128(&sPb[tid], bbox_pred + (size_t)(p0 + tid) * 4);
    async_g2lds_b32(&sLbl[tid], label_gt + tid);
    async_wait0();
    __syncthreads();

    // ---- WMMA one-hot GEMM: S[g][p] = cls_pred[p0+p][label[g]] ----
    // A (16x4 f32): lanes 0-15 -> K=k0,k0+1 ; lanes 16-31 -> K=k0+2,k0+3 (row M = lane&15)
    // B (4x16 f32): lane n = column; VGPR pair holds K=k0+{0,1} (lanes<16) / K=k0+{2,3} (>=16)
    // D (16x16 f32): VGPR r -> row r (lanes<16) / r+8 (lanes>=16), col = lane&15
    const int lane  = tid & 31;
    const int wid   = tid >> 5;
    const int halfk = (lane < 16) ? 0 : 2;
    const int nrow  = lane & 15;

    // per-lane labels for all 16 gt row-tiles (loop-invariant: hoist out of WMMA loop)
    int lbl16[16];
    #pragma unroll
    for (int gt = 0; gt < 16; ++gt) lbl16[gt] = sLbl[gt * 16 + nrow];

    for (int cc = 0; cc < 2; ++cc) {
        const int ct       = wid + cc * 8;        // column-tile 0..15
        const int colLocal = ct * 16 + nrow;
        const float* cp = cls_pred + (size_t)(p0 + colLocal) * CC;
        __builtin_prefetch(cp, 0, 0);             // global_prefetch_b8

        v2f b[20];                                 // K = 80 -> 20 k-steps of 4
        #pragma unroll
        for (int j = 0; j < 20; ++j) {
            b[j].x = cp[j * 4 + halfk];
            b[j].y = cp[j * 4 + halfk + 1];
        }
        #pragma unroll 1
        for (int gt = 0; gt < 16; ++gt) {
            const int lbl = lbl16[gt];
            v8f acc = {};
            #pragma unroll
            for (int j = 0; j < 20; ++j) {
                const int kb = j * 4 + halfk;
                v2f a;
                a.x = (lbl == kb)     ? 1.0f : 0.0f;
                a.y = (lbl == kb + 1) ? 1.0f : 0.0f;
                acc = __builtin_amdgcn_wmma_f32_16x16x4_f32(
                        false, a, false, b[j], (short)0, acc, false, false);
            }
            const int rowBase = gt * 16 + ((lane < 16) ? 0 : 8);
            #pragma unroll
            for (int r = 0; r < 8; ++r)
                sS[(rowBase + r) * SROW + colLocal] = acc[r];
        }
    }
    __syncthreads();

    // ---- per-thread (= per-gt) top-13 of align = score * iou^6 over this tile ----
    const float4 g4 = sGt[tid];
    const float  areaG = fmaxf(g4.z - g4.x, 0.0f) * fmaxf(g4.w - g4.y, 0.0f);

    float tv[TOPK];
    int   ti[TOPK];
    #pragma unroll
    for (int j = 0; j < TOPK; ++j) { tv[j] = -1.0f; ti[j] = 0; }

    for (int p = 0; p < TILE; ++p) {
        const float4 pb = sPb[p];                         // broadcast read
        float iw = fmaxf(fminf(g4.z, pb.z) - fmaxf(g4.x, pb.x), 0.0f);
        float ih = fmaxf(fminf(g4.w, pb.w) - fmaxf(g4.y, pb.y), 0.0f);
        float inter = iw * ih;
        float areaP = fmaxf(pb.z - pb.x, 0.0f) * fmaxf(pb.w - pb.y, 0.0f);
        float iou = inter / (areaG + areaP - inter + EPSV);
        float s  = sS[tid * SROW + p];                    // stride 257: conflict-free
        float i3 = iou * iou * iou;
        float al = s * i3 * i3;                           // score^1 * iou^6
        if (al > tv[TOPK - 1]) {
            // branchless bubble-through insert: constant indices keep arrays in VGPRs
            float cv_ = al; int ci_ = p0 + p;
            #pragma unroll
            for (int j = 0; j < TOPK; ++j) {
                bool sw = cv_ > tv[j];
                float tf = tv[j]; int tn = ti[j];
                tv[j] = sw ? cv_ : tv[j];
                ti[j] = sw ? ci_ : ti[j];
                cv_ = sw ? tf : cv_;
                ci_ = sw ? tn : ci_;
            }
        }
    }

    float* cv = candV + ((size_t)tid * NTILES + blockIdx.x) * TOPK;
    int*   ci = candI + ((size_t)tid * NTILES + blockIdx.x) * TOPK;
    #pragma unroll
    for (int j = 0; j < TOPK; ++j) { cv[j] = tv[j]; ci[j] = ti[j]; }
}

// ---------------------------------------------------------------------------
// stage 2: one block per gt: merge 256 partial top-13 lists -> global top-13,
// then scatter count/firstg (firstg == argmax of the mask column).
// ---------------------------------------------------------------------------
__global__ __launch_bounds__(256) void stage2_merge(
        const float* __restrict__ candV, const int* __restrict__ candI,
        int* __restrict__ counts, int* __restrict__ firstg) {
    __shared__ float v[CAND];
    __shared__ int   ix[CAND];
    __shared__ float rv[256];
    __shared__ int   rp[256];
    __shared__ int   chosen[TOPK];
    __shared__ int   flag;

    const int g = blockIdx.x, tid = threadIdx.x;
    const size_t base = ((size_t)g * NTILES + tid) * TOPK;
    #pragma unroll
    for (int j = 0; j < TOPK; ++j) {
        v[tid * TOPK + j]  = candV[base + j];
        ix[tid * TOPK + j] = candI[base + j];
    }
    __syncthreads();

    for (int round = 0; round < TOPK; ++round) {
        float mv = -3.0f; int mp = tid * TOPK;
        #pragma unroll
        for (int j = 0; j < TOPK; ++j) {
            float x = v[tid * TOPK + j];
            if (x > mv) { mv = x; mp = tid * TOPK + j; }
        }
        rv[tid] = mv; rp[tid] = mp;
        __syncthreads();
        for (int s = 128; s > 0; s >>= 1) {
            if (tid < s && rv[tid + s] > rv[tid]) {
                rv[tid] = rv[tid + s]; rp[tid] = rp[tid + s];
            }
            __syncthreads();
        }
        if (tid == 0) {
            if (round == 0) flag = (rv[0] > EPSV) ? 1 : 0;  // row-max gate
            chosen[round] = ix[rp[0]];
            v[rp[0]] = -2.0f;                               // remove winner
        }
        __syncthreads();
    }

    if (flag && tid < TOPK) {
        atomicAdd(&counts[chosen[tid]], 1);
        atomicMin(&firstg[chosen[tid]], g);
    }
}

// ---------------------------------------------------------------------------
// stage 3: per prior: resolve assignment and emit label / one-hot / bbox.
// One-hot written cooperatively: the block's 256x80 region is contiguous and
// 80 % 4 == 0, so float4 stores are fully coalesced and never straddle rows.
// ---------------------------------------------------------------------------
__global__ __launch_bounds__(256) void stage3_emit(
        const float* __restrict__ bbox_pred,
        const int*   __restrict__ label_gt,
        const float* __restrict__ bbox_gt,
        const int*   __restrict__ counts,
        const int*   __restrict__ firstg,
        const int*   __restrict__ bgp,
        float* __restrict__ out) {
    __shared__ float4 gbox[NN];
    __shared__ float  gArea[NN];
    __shared__ int    glbl[NN];
    __shared__ int    slbl[256];          // assigned label per local prior

    const int tid = threadIdx.x;
    async_g2lds_b128(&gbox[tid], bbox_gt + (size_t)tid * 4);
    async_g2lds_b32(&glbl[tid], label_gt + tid);
    async_wait0();
    __syncthreads();
    float4 gb = gbox[tid];
    gArea[tid] = fmaxf(gb.z - gb.x, 0.0f) * fmaxf(gb.w - gb.y, 0.0f);
    __syncthreads();

    const int p = blockIdx.x * 256 + tid;
    const float4 pb = ((const float4*)bbox_pred)[p];
    const float areaP = fmaxf(pb.z - pb.x, 0.0f) * fmaxf(pb.w - pb.y, 0.0f);

    const int c = counts[p];
    int g = 0;
    if (c == 1) {
        g = firstg[p];
    } else if (c > 1) {
        // conflicted prior: take first argmax_g iou(g, p) (strict '>' keeps first)
        float best = -1.0f;
        for (int gg = 0; gg < NN; ++gg) {
            float4 g4 = gbox[gg];
            float iw = fmaxf(fminf(g4.z, pb.z) - fmaxf(g4.x, pb.x), 0.0f);
            float ih = fmaxf(fminf(g4.w, pb.w) - fmaxf(g4.y, pb.y), 0.0f);
            float inter = iw * ih;
            float iou = inter / (gArea[gg] + areaP - inter + EPSV);
            if (iou > best) { best = iou; g = gg; }
        }
    }
    const int bg  = *bgp;
    const int lbl = (c > 0) ? glbl[g] : bg;
    slbl[tid] = lbl;

    // labels [P] and bboxes [P][4] (per-thread, coalesced)
    out[p] = (float)lbl;
    ((float4*)(out + PP + (size_t)PP * CC))[p] = gbox[g];   // all-zero column -> gt 0
    __syncthreads();

    // one-hot [P][80]: cooperative coalesced float4 stores over the block's region
    float4* crow = (float4*)(out + PP + (size_t)blockIdx.x * 256 * CC);
    #pragma unroll
    for (int it = 0; it < 20; ++it) {                 // 256*80/4 = 5120 = 20*256
        const int idx4 = tid + it * 256;
        const int pl   = idx4 / 20;                   // 20 float4 per prior row
        const int k4   = (idx4 - pl * 20) * 4;
        const int l    = slbl[pl];
        float4 vv;
        vv.x = (k4     == l) ? 1.0f : 0.0f;
        vv.y = (k4 + 1 == l) ? 1.0f : 0.0f;
        vv.z = (k4 + 2 == l) ? 1.0f : 0.0f;
        vv.w = (k4 + 3 == l) ? 1.0f : 0.0f;
        crow[idx4] = vv;
    }
}

// ---------------------------------------------------------------------------
extern "C" void kernel_launch(void* const* d_in, const int* in_sizes, int n_in,
                              void* d_out, int out_size, void* d_ws, size_t ws_size,
                              hipStream_t stream) {
    const float* cls_pred  = (const float*)d_in[0];
    const float* bbox_pred = (const float*)d_in[1];
    const int*   label_gt  = (const int*)d_in[2];
    const float* bbox_gt   = (const float*)d_in[3];
    const int*   bg_idx    = (const int*)d_in[4];
    float* out = (float*)d_out;

    // workspace layout (~7.3 MB)
    char* ws = (char*)d_ws;
    int*   counts = (int*)ws;                                  // P ints
    int*   firstg = (int*)(ws + (size_t)PP * 4);               // P ints
    float* candV  = (float*)(ws + (size_t)PP * 8);             // N*NTILES*TOPK floats
    int*   candI  = (int*)(ws + (size_t)PP * 8
                              + (size_t)NN * NTILES * TOPK * 4);

    stage0_init<<<PP / 256, 256, 0, stream>>>(counts, firstg);

    const size_t lds1 = (size_t)(NN * 4 + TILE * 4 + NN * SROW) * sizeof(float)
                      + (size_t)NN * sizeof(int);              // 272,384 B (< 320 KB/WGP)
    stage1_topk<<<NTILES, 256, lds1, stream>>>(cls_pred, bbox_pred, label_gt,
                                               bbox_gt, candV, candI);

    stage2_merge<<<NN, 256, 0, stream>>>(candV, candI, counts, firstg);

    stage3_emit<<<PP / 256, 256, 0, stream>>>(bbox_pred, label_gt, bbox_gt,
                                              counts, firstg, bg_idx, out);
}